// Resonance_17952963297933
// MI455X (gfx1250) — compile-verified
//
#include <hip/hip_runtime.h>
#include <math.h>

// Resonance synth:
//   freq[f]  = (MIN + SCALE*f0) * (f+1), zeroed if >= 1
//   env[f,t] = exp((t+1) * log(res[f] + 1e-12))            (t = frame 0..127)
//   out[s]   = sum_f lerp(env[f], frames->samples)(s) * sin(2*pi*(s+1)*freq[f])
//
// log-env table = rank-1 outer product -> computed with V_WMMA_F32_16X16X4_F32.

#define N_SAMPLES 32768
#define N_FRAMES  128
#define N_HARM    32
#define N_CH      64              // B*E = 4*16
#define SLICES    8               // sample slices per channel
#define SAMPLES_PER_SLICE (N_SAMPLES / SLICES)   // 4096
#define TWO_PI_F 6.28318530717958647692f

#define MIN_FREQ_F  (40.0f / 11025.0f)
#define FREQ_SCALE_F ((4000.0f - 40.0f) / 11025.0f)

typedef __attribute__((ext_vector_type(2))) float v2f;
typedef __attribute__((ext_vector_type(8))) float v8f;

__global__ __launch_bounds__(256) void resonance_kernel(
    const float* __restrict__ f0,       // (4,16,1) -> 64
    const float* __restrict__ res,      // (4,16,32) -> 2048
    const float* __restrict__ factors,  // 32 (1..32)
    float* __restrict__ out)            // (4,16,32768)
{
    __shared__ float s_env[N_HARM * N_FRAMES];   // env[f][t], 16 KB
    __shared__ float s_freq[N_HARM];
    __shared__ float s_logres[N_HARM];

    const int tid = threadIdx.x;
    const int ch  = blockIdx.x;          // channel = (b,e) pair, 0..63

    // ---- per-channel scalars -> LDS ----
    if (tid < N_HARM) {
        float fr0 = MIN_FREQ_F + FREQ_SCALE_F * f0[ch];
        float fr  = fr0 * factors[tid];
        s_freq[tid]   = (fr >= 1.0f) ? 0.0f : fr;         // aliased harmonics -> 0
        s_logres[tid] = logf(res[ch * N_HARM + tid] + 1e-12f);
    }
    __syncthreads();

    // ---- log-envelope outer product via WMMA, then exp -> LDS env table ----
    // Each of the 8 waves owns one 16-frame tile (t = 16*wave .. 16*wave+15)
    // and issues two 16x16x4 WMMAs (harmonic tiles f=0..15 and f=16..31).
    const int lane  = tid & 31;
    const int wave  = tid >> 5;
    const int tbase = wave * 16;

    // B (4x16): only K=0 row nonzero = frame index (t+1); lanes 0-15 / VGPR0.
    v2f bmat;
    bmat.x = (lane < 16) ? (float)(tbase + lane + 1) : 0.0f;
    bmat.y = 0.0f;

    const int tcol = tbase + (lane & 15);
    const int fhi  = (lane >= 16) ? 8 : 0;

#pragma unroll
    for (int ft = 0; ft < 2; ++ft) {
        // A (16x4): only K=0 column nonzero = logres[f]; lanes 0-15 / VGPR0.
        v2f amat;
        amat.x = (lane < 16) ? s_logres[ft * 16 + lane] : 0.0f;
        amat.y = 0.0f;

        v8f d;
#if __has_builtin(__builtin_amdgcn_wmma_f32_16x16x4_f32)
        v8f c = {};
        // (neg_a, A, neg_b, B, c_mod, C, reuse_a, reuse_b)
        d = __builtin_amdgcn_wmma_f32_16x16x4_f32(
                false, amat, false, bmat, (short)0, c, false, false);
#else
        // scalar fallback: d[j] = logres[f] * (t+1)
#pragma unroll
        for (int j = 0; j < 8; ++j) {
            int fl = ft * 16 + j + fhi;
            d[j] = s_logres[fl] * (float)(tcol + 1);
        }
#endif
        // D layout: VGPR j -> M = j (lanes 0-15) or j+8 (lanes 16-31), N = lane%16
#pragma unroll
        for (int j = 0; j < 8; ++j) {
            int f = ft * 16 + j + fhi;
            s_env[f * N_FRAMES + tcol] = __expf(d[j]);
        }
    }
    __syncthreads();

    // ---- sample loop: lin_interp(env) * sin(2*pi*(s+1)*freq), sum over f ----
    float* outc = out + (size_t)ch * N_SAMPLES;
    const int s0 = (int)blockIdx.y * SAMPLES_PER_SLICE;

    for (int s = s0 + tid; s < s0 + SAMPLES_PER_SLICE; s += 256) {
        // lin_interp coords: (s+0.5)/256 - 0.5, clipped to [0,127]
        float coords = (s + 0.5f) * (1.0f / 256.0f) - 0.5f;
        coords = fminf(fmaxf(coords, 0.0f), 127.0f);
        int   i0 = (int)coords;
        int   i1 = min(i0 + 1, N_FRAMES - 1);
        float w  = coords - (float)i0;

        float sum = 0.0f;
#pragma unroll 8
        for (int f = 0; f < N_HARM; ++f) {
            float fr = s_freq[f];
            // phase range reduction in f64: (s+1)*fr can be ~3e4 cycles,
            // fp32 alone would lose the fractional turn entirely.
            double p    = (double)(s + 1) * (double)fr;
            float  frac = (float)(p - floor(p));          // in [0,1)
            float  e0   = s_env[f * N_FRAMES + i0];
            float  e1   = s_env[f * N_FRAMES + i1];
            float  env  = fmaf(e1 - e0, w, e0);
            sum = fmaf(env, __sinf(TWO_PI_F * frac), sum);
        }
        outc[s] = sum;
    }
}

extern "C" void kernel_launch(void* const* d_in, const int* in_sizes, int n_in,
                              void* d_out, int out_size, void* d_ws, size_t ws_size,
                              hipStream_t stream) {
    const float* f0      = (const float*)d_in[0];
    const float* res     = (const float*)d_in[1];
    const float* factors = (const float*)d_in[2];
    float* out = (float*)d_out;

    dim3 grid(N_CH, SLICES);   // 64 channels x 8 sample slices = 512 blocks
    resonance_kernel<<<grid, 256, 0, stream>>>(f0, res, factors, out);
}